// MultiHeadCrossAttention_81295140979030
// MI455X (gfx1250) — compile-verified
//
#include <hip/hip_runtime.h>
#include <hip/hip_bf16.h>

// MI455X / gfx1250, wave32. All-fp32 pipeline using V_WMMA_F32_16X16X4_F32.
//
// scores[b,l,n] = (Q[b,l]·K[b,l,n]) / (H*sqrt(D))
//   with Q = state Wq + bq, K = A Wk + bk
// = ( state·(Wq Wk^T)·A^T + state·(Wq bk) + A·(Wk bq) + bq·bk ) / (H*sqrt(D))
//
// Pipeline:
//   K0 bias_prep : M[:,1024] = Wq bk (v),  M[:,1025..1039]=0,
//                  u[e<1024] = Wk bq, u[1024] = bq·bk, u[1025..]=0
//   K1 gemm (NT) : M[:, :1024] = Wq (2048x1024) x Wk^T (1024x1024)
//   K2 gemm (NN) : QM (8192x1040) = state (8192x2048) x M (2048x1040)  + u
//   K3 scores    : out[bl,n] = (QM[bl,:1024]·A[bl,n,:] + QM[bl,1024]) * INV

typedef float v2f __attribute__((ext_vector_type(2)));
typedef float v4f __attribute__((ext_vector_type(4)));
typedef float v8f __attribute__((ext_vector_type(8)));

#define LDM   1040   // 1024 cols + 1 bias col + 15 pad (65*16)
#define TILE_R 128
#define TILE_C 128
#define TILE_K 32
#define AT_STRIDE 36   // 32 + 4 pad : conflict-free for lane-row fragment reads
#define BT_STRIDE 36   // bT[n][k], k-contiguous: fragment = one aligned b64 read

// ---------------------------------------------------------------------------
// K0: bias columns.  i<2048: v-row into M[:,1024..1039]; else u vector.
__global__ void bias_prep_kernel(const float* __restrict__ Wq,
                                 const float* __restrict__ bq,
                                 const float* __restrict__ Wk,
                                 const float* __restrict__ bk,
                                 float* __restrict__ Mbuf,
                                 float* __restrict__ ubuf) {
    int i = blockIdx.x * blockDim.x + threadIdx.x;
    if (i < 2048) {
        const float* w = Wq + (size_t)i * 1024;
        float s = 0.f;
        for (int f = 0; f < 1024; ++f) s += w[f] * bk[f];
        float* row = Mbuf + (size_t)i * LDM;
        row[1024] = s;
        for (int j = 1025; j < LDM; ++j) row[j] = 0.f;
    } else if (i < 2048 + LDM) {
        int e = i - 2048;
        if (e < 1024) {
            const float* w = Wk + (size_t)e * 1024;
            float s = 0.f;
            for (int f = 0; f < 1024; ++f) s += w[f] * bq[f];
            ubuf[e] = s;
        } else if (e == 1024) {
            float s = 0.f;
            for (int f = 0; f < 1024; ++f) s += bq[f] * bk[f];
            ubuf[e] = s;
        } else {
            ubuf[e] = 0.f;
        }
    }
}

// ---------------------------------------------------------------------------
// Tiled fp32 WMMA GEMM: C[rows x cols] = A[rows x k] * op(B) (+ bias[col]).
// transB=0: B is k x cols (row-major).  transB=1: B is cols x k (row-major),
// i.e. C = A * B^T (NT gemm).  rows % 128 == 0, k % 32 == 0 required.
// Fragment layouts per CDNA5 ISA 7.12.2:
//   A 16x4 f32 : lane l holds A[l%16, 2*(l/16)+j], j = vgpr 0,1
//   B  4x16 f32: lane l holds B[2*(l/16)+j, l%16]
//   C 16x16 f32: lane l vgpr i holds C[i + 8*(l/16), l%16]
// LDS: aT[row][k] (stride 36), bT[col][k] (stride 36) -> both fragments are
// single contiguous 8-byte LDS reads (ds_load_*_b64-mergeable, no repacking).
__global__ __launch_bounds__(256)
void gemm_f32_wmma_kernel(const float* __restrict__ A, int lda,
                          const float* __restrict__ B, int ldb, int transB,
                          float* __restrict__ C, int ldc,
                          int rows, int cols, int kdim,
                          const float* __restrict__ bias) {
    __shared__ float aT[TILE_R * AT_STRIDE];
    __shared__ float bT[TILE_C * BT_STRIDE];

    const int nColTiles = (cols + TILE_C - 1) / TILE_C;
    const int tileC = (blockIdx.x % nColTiles) * TILE_C;
    const int tileR = (blockIdx.x / nColTiles) * TILE_R;

    const int lane  = threadIdx.x & 31;
    const int wv    = threadIdx.x >> 5;   // 8 waves: wave wv owns rows [16*wv,16*wv+16)
    const int rloc  = lane & 15;
    const int khalf = lane >> 4;          // k offset 2*khalf inside WMMA fragment

    v8f acc[8] = {};                      // 16 rows x 128 cols per wave

    for (int s0 = 0; s0 < kdim; s0 += TILE_K) {
        // ---- stage A block: aT[r][k], rows [tileR,+128), k [s0,+32) --------
        {
            int r = threadIdx.x >> 1;
            int h = (threadIdx.x & 1) * 16;
            const v4f* src =
                reinterpret_cast<const v4f*>(A + (size_t)(tileR + r) * lda + s0 + h);
            v4f* dst = reinterpret_cast<v4f*>(&aT[r * AT_STRIDE + h]);
#pragma unroll
            for (int q = 0; q < 4; ++q) dst[q] = src[q];
        }
        // ---- stage B block: bT[n][k] = op(B)[s0+k][tileC+n] ----------------
        if (!transB) {
            // B row-major k x cols: coalesced global reads, LDS scatter stores
            int r   = threadIdx.x >> 3;            // 0..31 (k)
            int cch = (threadIdx.x & 7) * 16;      // col chunk
            const float* src = B + (size_t)(s0 + r) * ldb + tileC + cch;
#pragma unroll
            for (int q = 0; q < 4; ++q) {
                v4f v = {0.f, 0.f, 0.f, 0.f};
                if (tileC + cch + q * 4 < cols)
                    v = *reinterpret_cast<const v4f*>(src + q * 4);
                bT[(cch + q * 4 + 0) * BT_STRIDE + r] = v.x;
                bT[(cch + q * 4 + 1) * BT_STRIDE + r] = v.y;
                bT[(cch + q * 4 + 2) * BT_STRIDE + r] = v.z;
                bT[(cch + q * 4 + 3) * BT_STRIDE + r] = v.w;
            }
        } else {
            // B row-major cols x k: k contiguous -> fully vectorized staging
            int n = threadIdx.x >> 1;              // 0..127 (col)
            int h = (threadIdx.x & 1) * 16;        // k half
            int col = tileC + n;
#pragma unroll
            for (int q = 0; q < 4; ++q) {
                v4f v = {0.f, 0.f, 0.f, 0.f};
                if (col < cols)
                    v = *reinterpret_cast<const v4f*>(
                        B + (size_t)col * ldb + s0 + h + q * 4);
                *reinterpret_cast<v4f*>(&bT[n * BT_STRIDE + h + q * 4]) = v;
            }
        }
        __syncthreads();

        // ---- WMMA inner loop ----------------------------------------------
#pragma unroll
        for (int ks = 0; ks < TILE_K; ks += 4) {
            v2f a = *reinterpret_cast<const v2f*>(
                &aT[(wv * 16 + rloc) * AT_STRIDE + ks + 2 * khalf]);
#pragma unroll
            for (int sub = 0; sub < 8; ++sub) {
                v2f b = *reinterpret_cast<const v2f*>(
                    &bT[(sub * 16 + rloc) * BT_STRIDE + ks + 2 * khalf]);
                acc[sub] = __builtin_amdgcn_wmma_f32_16x16x4_f32(
                    false, a, false, b, (short)0, acc[sub], false, false);
            }
        }
        __syncthreads();
    }

    // ---- epilogue: C-layout store (+ optional bias) ------------------------
#pragma unroll
    for (int sub = 0; sub < 8; ++sub) {
        int col = tileC + sub * 16 + rloc;
        if (col < cols) {
            float badd = bias ? bias[col] : 0.f;
#pragma unroll
            for (int i = 0; i < 8; ++i) {
                int row = tileR + wv * 16 + i + 8 * khalf;
                C[(size_t)row * ldc + col] = acc[sub][i] + badd;
            }
        }
    }
}

// ---------------------------------------------------------------------------
// K3: one workgroup per (b,l).  16 dots of length 1024 against the LDS-staged
// QM row; action_embs streamed once with nontemporal v4f loads.
__global__ __launch_bounds__(256)
void scores_kernel(const float* __restrict__ A,    // action_embs (BL,16,1024)
                   const float* __restrict__ QM,   // (BL, LDM)
                   float* __restrict__ out) {      // (BL, 16)
    const float INV = 0.011048543456039806f;       // 1 / (8 * sqrt(128))
    __shared__ __align__(16) float q[LDM];

    const int bl = blockIdx.x;
    const v4f* qmsrc = reinterpret_cast<const v4f*>(QM + (size_t)bl * LDM);
    reinterpret_cast<v4f*>(q)[threadIdx.x] = qmsrc[threadIdx.x];      // 1024 floats
    if (threadIdx.x == 0) q[1024] = QM[(size_t)bl * LDM + 1024];      // c term
    __syncthreads();

    const int lane = threadIdx.x & 31;
    const int wv   = threadIdx.x >> 5;

#pragma unroll
    for (int nn = 0; nn < 2; ++nn) {
        int n = wv * 2 + nn;
        const v4f* a4 =
            reinterpret_cast<const v4f*>(A + ((size_t)bl * 16 + n) * 1024);
        float s = 0.f;
#pragma unroll
        for (int i = 0; i < 8; ++i) {
            v4f av = __builtin_nontemporal_load(a4 + i * 32 + lane);
            v4f qv = reinterpret_cast<const v4f*>(q)[i * 32 + lane];
            s += av.x * qv.x + av.y * qv.y + av.z * qv.z + av.w * qv.w;
        }
#pragma unroll
        for (int off = 16; off > 0; off >>= 1) s += __shfl_xor(s, off, 32);
        if (lane == 0) out[(size_t)bl * 16 + n] = (s + q[1024]) * INV;
    }
}

// ---------------------------------------------------------------------------
extern "C" void kernel_launch(void* const* d_in, const int* in_sizes, int n_in,
                              void* d_out, int out_size, void* d_ws, size_t ws_size,
                              hipStream_t stream) {
    (void)in_sizes; (void)n_in; (void)out_size; (void)ws_size;
    const float* state = (const float*)d_in[0];   // (4,2048,2048)
    const float* aemb  = (const float*)d_in[1];   // (4,2048,16,1024)
    const float* Wq    = (const float*)d_in[2];   // (2048,1024)
    const float* bq    = (const float*)d_in[3];   // (1024,)
    const float* Wk    = (const float*)d_in[4];   // (1024,1024)
    const float* bk    = (const float*)d_in[5];   // (1024,)
    float* out = (float*)d_out;

    const int BL = 4 * 2048;                      // 8192
    char* ws = (char*)d_ws;
    float* Mbuf = (float*)ws;                                   // 2048*1040*4  = 8,519,680 B
    float* ubuf = (float*)(ws + (size_t)2048 * LDM * 4);        // 1040*4       = 4,160 B
    float* QM   = (float*)(ws + (size_t)2048 * LDM * 4 + 4160); // 8192*1040*4  = 34,078,720 B

    // K0: bias columns of M and u vector (exact zeros here since bq=bk=0)
    bias_prep_kernel<<<(2048 + LDM + 255) / 256, 256, 0, stream>>>(
        Wq, bq, Wk, bk, Mbuf, ubuf);

    // K1: M[:, :1024] = Wq x Wk^T   (2048x1024, K=1024, NT)
    {
        int rows = 2048, cols = 1024, k = 1024;
        int grid = (rows / TILE_R) * ((cols + TILE_C - 1) / TILE_C);  // 16*8=128
        gemm_f32_wmma_kernel<<<grid, 256, 0, stream>>>(
            Wq, 1024, Wk, 1024, /*transB=*/1, Mbuf, LDM, rows, cols, k, nullptr);
    }

    // K2: QM = state x M + u   (8192x1040, K=2048, NN)
    {
        int rows = BL, cols = LDM, k = 2048;
        int grid = (rows / TILE_R) * ((cols + TILE_C - 1) / TILE_C);  // 64*9=576
        gemm_f32_wmma_kernel<<<grid, 256, 0, stream>>>(
            state, 2048, Mbuf, LDM, /*transB=*/0, QM, LDM, rows, cols, k, ubuf);
    }

    // K3: streaming batched dot -> scores
    scores_kernel<<<BL, 256, 0, stream>>>(aemb, QM, out);
}